// MILNCEHead_42262478193385
// MI455X (gfx1250) — compile-verified
//
#include <hip/hip_runtime.h>

typedef __attribute__((ext_vector_type(2))) float v2f;
typedef __attribute__((ext_vector_type(8))) float v8f;

#define N_VID   4096
#define T_CLIP  4
#define NT      (N_VID * T_CLIP)     // 16384
#define D_FEAT  256
#define INV_TEMP 10.0f

// workspace layout (32-bit words)
#define OFF_ROWMAX 0                          // uint[4096]
#define OFF_COLMAX (OFF_ROWMAX + N_VID)       // uint[16384]
#define OFF_COUNTS (OFF_COLMAX + NT)          // int[16384]
#define OFF_POS    (OFF_COUNTS + NT)          // float[16384]
#define OFF_NOM    (OFF_POS + NT)             // float[4096]
#define OFF_DENMAX (OFF_NOM + N_VID)          // float[4096]
#define OFF_DENACC (OFF_DENMAX + N_VID)       // float[4096]
#define OFF_PART   (OFF_DENACC + N_VID)       // float[16*5]
#define WS_WORDS   (OFF_PART + 128)

// ds_swizzle group-of-32 XOR pattern: offset = (xor<<10) | (or<<5) | and(0x1f)
#define SWZ_XOR(x, m) \
    __int_as_float(__builtin_amdgcn_ds_swizzle(__float_as_int(x), ((m) << 10) | 0x1f))

// order-preserving float <-> uint key for atomic max
__device__ __forceinline__ unsigned fkey(float f) {
    unsigned b = __float_as_uint(f);
    return (b & 0x80000000u) ? ~b : (b | 0x80000000u);
}
__device__ __forceinline__ float fdec(unsigned k) {
    return (k & 0x80000000u) ? __uint_as_float(k & 0x7FFFFFFFu) : __uint_as_float(~k);
}

// max-reduce across each 16-lane half (xor masks stay inside the half)
__device__ __forceinline__ float redmax16(float x) {
    x = fmaxf(x, SWZ_XOR(x, 1));
    x = fmaxf(x, SWZ_XOR(x, 2));
    x = fmaxf(x, SWZ_XOR(x, 4));
    x = fmaxf(x, SWZ_XOR(x, 8));
    return x;
}
__device__ __forceinline__ float redadd16(float x) {
    x += SWZ_XOR(x, 1);
    x += SWZ_XOR(x, 2);
    x += SWZ_XOR(x, 4);
    x += SWZ_XOR(x, 8);
    return x;
}

// ---------------- init ----------------
__global__ void k_init(unsigned* ws, float* out) {
    int i = blockIdx.x * blockDim.x + threadIdx.x;
    if (i < WS_WORDS) ws[i] = 0u;
    if (i < 5) out[i] = 0.0f;
}

// single 16x16 fp32 WMMA tile, K-loop over 256 (deterministic: same
// (rowBase,colBase) always gives bit-identical results)
__device__ __forceinline__ v8f wmma_tile(const float* __restrict__ v,
                                         const float* __restrict__ t,
                                         int rowBase, int colBase, int lane) {
    const int half = lane >> 4;
    const int l = lane & 15;
    const size_t aRow = (size_t)(rowBase + l) * D_FEAT + 2 * half;
    const size_t bRow = (size_t)(colBase + l) * D_FEAT + 2 * half;
    v8f acc = {};
#pragma unroll 8
    for (int k = 0; k < D_FEAT; k += 4) {
        v2f a, b;
        a.x = v[aRow + k];
        a.y = v[aRow + k + 1];
        b.x = t[bRow + k];
        b.y = t[bRow + k + 1];
        acc = __builtin_amdgcn_wmma_f32_16x16x4_f32(false, a, false, b,
                                                    (short)0, acc, false, false);
    }
    return acc;
}

// ------------- positives via WMMA (bit-identical to pass1 tiles) -------------
// diag block of rows [r, r+16) lives in columns [4r, 4r+64) = 4 WMMA tiles.
// block = 8 waves: 2 row-groups x 4 column tiles.
__global__ __launch_bounds__(256)
void k_diagwmma(const float* __restrict__ v, const float* __restrict__ t,
                float* __restrict__ pos) {
    const int lane = threadIdx.x & 31;
    const int wave = threadIdx.x >> 5;
    const int rowBase = blockIdx.x * 32 + (wave >> 2) * 16;
    const int colBase = rowBase * 4 + (wave & 3) * 16;
    const int half = lane >> 4;
    const int l = lane & 15;

    v8f acc = wmma_tile(v, t, rowBase, colBase, lane);

    const int c = colBase + l;
#pragma unroll
    for (int j = 0; j < 8; ++j) {
        const int m = rowBase + half * 8 + j;
        if ((c >> 2) == m) pos[c] = acc[j] * INV_TEMP;
    }
}

// ------------- nominator: logsumexp over the 4 positives -------------
__global__ void k_nom(const float* __restrict__ pos, float* __restrict__ nom) {
    int i = blockIdx.x * blockDim.x + threadIdx.x;
    if (i >= N_VID) return;
    float p0 = pos[i * 4 + 0], p1 = pos[i * 4 + 1];
    float p2 = pos[i * 4 + 2], p3 = pos[i * 4 + 3];
    float mx = fmaxf(fmaxf(p0, p1), fmaxf(p2, p3));
    float se = __expf(p0 - mx) + __expf(p1 - mx) + __expf(p2 - mx) + __expf(p3 - mx);
    nom[i] = mx + __logf(se);
}

// 16x64 register-blocked GEMM: one A fragment feeds 4 WMMA tiles
#define GEMM_PROLOG()                                                          \
    const int lane = threadIdx.x & 31;                                         \
    const int wave = threadIdx.x >> 5;                                         \
    const int rowBase = blockIdx.y * 64 + (wave >> 1) * 16;                    \
    const int colWave = blockIdx.x * 128 + (wave & 1) * 64;                    \
    const int half = lane >> 4;                                                \
    const int l = lane & 15;                                                   \
    const size_t aRow = (size_t)(rowBase + l) * D_FEAT + 2 * half;             \
    const size_t bR0 = (size_t)(colWave + l) * D_FEAT + 2 * half;              \
    const size_t bR1 = bR0 + (size_t)16 * D_FEAT;                              \
    const size_t bR2 = bR0 + (size_t)32 * D_FEAT;                              \
    const size_t bR3 = bR0 + (size_t)48 * D_FEAT;                              \
    v8f acc0 = {}, acc1 = {}, acc2 = {}, acc3 = {};                            \
    _Pragma("unroll 8")                                                        \
    for (int k = 0; k < D_FEAT; k += 4) {                                      \
        v2f a, b0, b1, b2, b3;                                                 \
        a.x = v[aRow + k];      a.y = v[aRow + k + 1];                         \
        b0.x = t[bR0 + k];      b0.y = t[bR0 + k + 1];                         \
        b1.x = t[bR1 + k];      b1.y = t[bR1 + k + 1];                         \
        b2.x = t[bR2 + k];      b2.y = t[bR2 + k + 1];                         \
        b3.x = t[bR3 + k];      b3.y = t[bR3 + k + 1];                         \
        acc0 = __builtin_amdgcn_wmma_f32_16x16x4_f32(false, a, false, b0,      \
                                                     (short)0, acc0, false, false); \
        acc1 = __builtin_amdgcn_wmma_f32_16x16x4_f32(false, a, false, b1,      \
                                                     (short)0, acc1, false, false); \
        acc2 = __builtin_amdgcn_wmma_f32_16x16x4_f32(false, a, false, b2,      \
                                                     (short)0, acc2, false, false); \
        acc3 = __builtin_amdgcn_wmma_f32_16x16x4_f32(false, a, false, b3,      \
                                                     (short)0, acc3, false, false); \
    }

// ------------- pass 1: row/col maxima + rank counts -------------
__global__ __launch_bounds__(256)
void k_pass1(const float* __restrict__ v, const float* __restrict__ t,
             const float* __restrict__ pos,
             unsigned* __restrict__ rowMaxKey, unsigned* __restrict__ colMaxKey,
             int* __restrict__ counts) {
    GEMM_PROLOG()

    float cm0 = -3.4e38f, cm1 = -3.4e38f, cm2 = -3.4e38f, cm3 = -3.4e38f;
#pragma unroll
    for (int j = 0; j < 8; ++j) {
        float v0 = acc0[j] * INV_TEMP;
        float v1 = acc1[j] * INV_TEMP;
        float v2 = acc2[j] * INV_TEMP;
        float v3 = acc3[j] * INV_TEMP;
        cm0 = fmaxf(cm0, v0); cm1 = fmaxf(cm1, v1);
        cm2 = fmaxf(cm2, v2); cm3 = fmaxf(cm3, v3);
        const int m = rowBase + half * 8 + j;
        float rmax = redmax16(fmaxf(fmaxf(v0, v1), fmaxf(v2, v3)));
        if (l == 0) atomicMax(&rowMaxKey[m], fkey(rmax));
#pragma unroll
        for (int tt = 0; tt < T_CLIP; ++tt) {
            float thr = pos[m * T_CLIP + tt];
            unsigned g0 = (unsigned)__ballot(v0 > thr);
            unsigned g1 = (unsigned)__ballot(v1 > thr);
            unsigned g2 = (unsigned)__ballot(v2 > thr);
            unsigned g3 = (unsigned)__ballot(v3 > thr);
            if (l == 0) {
                int cnt = half
                    ? (__popc(g0 >> 16) + __popc(g1 >> 16) + __popc(g2 >> 16) + __popc(g3 >> 16))
                    : (__popc(g0 & 0xFFFFu) + __popc(g1 & 0xFFFFu) +
                       __popc(g2 & 0xFFFFu) + __popc(g3 & 0xFFFFu));
                if (cnt) atomicAdd(&counts[m * T_CLIP + tt], cnt);
            }
        }
    }
    atomicMax(&colMaxKey[colWave + l],      fkey(cm0));
    atomicMax(&colMaxKey[colWave + 16 + l], fkey(cm1));
    atomicMax(&colMaxKey[colWave + 32 + l], fkey(cm2));
    atomicMax(&colMaxKey[colWave + 48 + l], fkey(cm3));
}

// ------------- denominator max combine -------------
__global__ void k_denmax(const unsigned* __restrict__ rowMaxKey,
                         const unsigned* __restrict__ colMaxKey,
                         float* __restrict__ denMax) {
    int i = blockIdx.x * blockDim.x + threadIdx.x;
    if (i >= N_VID) return;
    float m = fdec(rowMaxKey[i]);
#pragma unroll
    for (int tt = 0; tt < T_CLIP; ++tt)
        m = fmaxf(m, fdec(colMaxKey[i * T_CLIP + tt]));
    denMax[i] = m;
}

// ------------- pass 2: exp-sums for denominator -------------
__global__ __launch_bounds__(256)
void k_pass2(const float* __restrict__ v, const float* __restrict__ t,
             const float* __restrict__ denMax, float* __restrict__ denAcc) {
    GEMM_PROLOG()

    const int cv0 = (colWave + l) >> 2;
    const int cv1 = (colWave + 16 + l) >> 2;
    const int cv2 = (colWave + 32 + l) >> 2;
    const int cv3 = (colWave + 48 + l) >> 2;
    const float d0 = denMax[cv0], d1 = denMax[cv1];
    const float d2 = denMax[cv2], d3 = denMax[cv3];
    float cs0 = 0.f, cs1 = 0.f, cs2 = 0.f, cs3 = 0.f;
#pragma unroll
    for (int j = 0; j < 8; ++j) {
        float v0 = acc0[j] * INV_TEMP;
        float v1 = acc1[j] * INV_TEMP;
        float v2 = acc2[j] * INV_TEMP;
        float v3 = acc3[j] * INV_TEMP;
        const int m = rowBase + half * 8 + j;
        const float dm = denMax[m];
        float rs = __expf(v0 - dm) + __expf(v1 - dm) +
                   __expf(v2 - dm) + __expf(v3 - dm);
        rs = redadd16(rs);
        if (l == 0) atomicAdd(&denAcc[m], rs);
        cs0 += __expf(v0 - d0); cs1 += __expf(v1 - d1);
        cs2 += __expf(v2 - d2); cs3 += __expf(v3 - d3);
    }
    atomicAdd(&denAcc[cv0], cs0);
    atomicAdd(&denAcc[cv1], cs1);
    atomicAdd(&denAcc[cv2], cs2);
    atomicAdd(&denAcc[cv3], cs3);
}

// ------------- per-video metrics + block partials -------------
__global__ __launch_bounds__(256)
void k_final(const float* __restrict__ denMax, const float* __restrict__ denAcc,
             const float* __restrict__ nom, const int* __restrict__ counts,
             float* __restrict__ partials) {
    __shared__ float sm[256];
    int i = blockIdx.x * 256 + threadIdx.x;
    float den = denMax[i] + __logf(denAcc[i]);
    float loss = den - nom[i];
    int r0 = counts[i * 4 + 0], r1 = counts[i * 4 + 1];
    int r2 = counts[i * 4 + 2], r3 = counts[i * 4 + 3];
    float vals[5];
    vals[0] = loss;
    vals[1] = ((r0 == 0) + (r1 == 0) + (r2 == 0) + (r3 == 0)) * 0.25f;
    vals[2] = ((r0 < 5) + (r1 < 5) + (r2 < 5) + (r3 < 5)) * 0.25f;
    vals[3] = ((r0 < 10) + (r1 < 10) + (r2 < 10) + (r3 < 10)) * 0.25f;
    vals[4] = (float)(r0 + r1 + r2 + r3) * 0.25f;
#pragma unroll
    for (int q = 0; q < 5; ++q) {
        sm[threadIdx.x] = vals[q];
        __syncthreads();
        for (int s = 128; s > 0; s >>= 1) {
            if (threadIdx.x < s) sm[threadIdx.x] += sm[threadIdx.x + s];
            __syncthreads();
        }
        if (threadIdx.x == 0) partials[blockIdx.x * 5 + q] = sm[0];
        __syncthreads();
    }
}

__global__ void k_reduce2(const float* __restrict__ partials, float* __restrict__ out) {
    int q = threadIdx.x;
    if (q < 5) {
        float s = 0.0f;
        for (int b = 0; b < N_VID / 256; ++b) s += partials[b * 5 + q];
        out[q] = s * (1.0f / N_VID);
    }
}

extern "C" void kernel_launch(void* const* d_in, const int* in_sizes, int n_in,
                              void* d_out, int out_size, void* d_ws, size_t ws_size,
                              hipStream_t stream) {
    const float* v = (const float*)d_in[0];   // [4096, 256]
    const float* t = (const float*)d_in[1];   // [16384, 256]
    float* out = (float*)d_out;               // 5 scalars
    unsigned* ws = (unsigned*)d_ws;

    unsigned* rowMaxKey = ws + OFF_ROWMAX;
    unsigned* colMaxKey = ws + OFF_COLMAX;
    int*      counts    = (int*)(ws + OFF_COUNTS);
    float*    pos       = (float*)(ws + OFF_POS);
    float*    nom       = (float*)(ws + OFF_NOM);
    float*    denMax    = (float*)(ws + OFF_DENMAX);
    float*    denAcc    = (float*)(ws + OFF_DENACC);
    float*    partials  = (float*)(ws + OFF_PART);

    k_init<<<(WS_WORDS + 255) / 256, 256, 0, stream>>>(ws, out);
    k_diagwmma<<<N_VID / 32, 256, 0, stream>>>(v, t, pos);
    k_nom<<<N_VID / 256, 256, 0, stream>>>(pos, nom);

    dim3 g(NT / 128, N_VID / 64);  // block tile: 64 rows x 128 cols, 8 waves
    k_pass1<<<g, 256, 0, stream>>>(v, t, pos, rowMaxKey, colMaxKey, counts);
    k_denmax<<<N_VID / 256, 256, 0, stream>>>(rowMaxKey, colMaxKey, denMax);
    k_pass2<<<g, 256, 0, stream>>>(v, t, denMax, denAcc);

    k_final<<<N_VID / 256, 256, 0, stream>>>(denMax, denAcc, nom, counts, partials);
    k_reduce2<<<1, 32, 0, stream>>>(partials, out);
}